// BasicConvolutionBlock_40759239639902
// MI455X (gfx1250) — compile-verified
//
#include <hip/hip_runtime.h>
#include <hip/hip_bf16.h>

typedef __attribute__((ext_vector_type(2))) float v2f;
typedef __attribute__((ext_vector_type(8))) float v8f;

#define CIN   64
#define COUT  64
#define NCELLS (2u * 128u * 128u * 128u)   // 4,194,304 = 2^22
#define SCAN_BLOCK 256
#define SCAN_ITEMS 16
#define SCAN_ELEMS (SCAN_BLOCK * SCAN_ITEMS)   // 4096
#define SCAN_NBLK  (NCELLS / SCAN_ELEMS)       // 1024

// ---------------------------------------------------------------------------
// 1) compute dense-grid cell per point, mark occupancy flags
// ---------------------------------------------------------------------------
__global__ void sc_keys_kernel(const int* __restrict__ coords,
                               unsigned* __restrict__ flags,
                               unsigned* __restrict__ pointCell, int N) {
    int i = blockIdx.x * blockDim.x + threadIdx.x;
    if (i >= N) return;
    int x = coords[i * 4 + 0] >> 1;   // stride-2 downsample (coords >= 0)
    int y = coords[i * 4 + 1] >> 1;
    int z = coords[i * 4 + 2] >> 1;
    int b = coords[i * 4 + 3];
    unsigned cell = ((unsigned)b << 21) | ((unsigned)x << 14) |
                    ((unsigned)y << 7) | (unsigned)z;
    pointCell[i] = cell;
    flags[cell] = 1u;     // benign race: all write 1
}

// ---------------------------------------------------------------------------
// 2) three-kernel exclusive scan of flags -> ranks
// ---------------------------------------------------------------------------
__global__ void sc_scan1_kernel(const unsigned* __restrict__ in,
                                unsigned* __restrict__ out,
                                unsigned* __restrict__ blockSums) {
    __shared__ unsigned sh[SCAN_BLOCK];
    unsigned base = blockIdx.x * SCAN_ELEMS + threadIdx.x * SCAN_ITEMS;
    unsigned local[SCAN_ITEMS];
    unsigned sum = 0;
    #pragma unroll
    for (int i = 0; i < SCAN_ITEMS; ++i) { local[i] = sum; sum += in[base + i]; }
    sh[threadIdx.x] = sum;
    __syncthreads();
    for (int off = 1; off < SCAN_BLOCK; off <<= 1) {
        unsigned v = 0;
        if ((int)threadIdx.x >= off) v = sh[threadIdx.x - off];
        __syncthreads();
        sh[threadIdx.x] += v;
        __syncthreads();
    }
    unsigned tOff = (threadIdx.x == 0) ? 0u : sh[threadIdx.x - 1];
    #pragma unroll
    for (int i = 0; i < SCAN_ITEMS; ++i) out[base + i] = tOff + local[i];
    if (threadIdx.x == SCAN_BLOCK - 1) blockSums[blockIdx.x] = sh[SCAN_BLOCK - 1];
}

__global__ void sc_scan2_kernel(unsigned* __restrict__ blockSums,
                                unsigned* __restrict__ nUnique) {
    // exclusive scan of SCAN_NBLK (=1024) block sums; 256 threads x 4 items
    __shared__ unsigned sh[SCAN_BLOCK];
    unsigned local[4];
    unsigned sum = 0;
    int base = threadIdx.x * 4;
    #pragma unroll
    for (int i = 0; i < 4; ++i) { local[i] = sum; sum += blockSums[base + i]; }
    sh[threadIdx.x] = sum;
    __syncthreads();
    for (int off = 1; off < SCAN_BLOCK; off <<= 1) {
        unsigned v = 0;
        if ((int)threadIdx.x >= off) v = sh[threadIdx.x - off];
        __syncthreads();
        sh[threadIdx.x] += v;
        __syncthreads();
    }
    unsigned tOff = (threadIdx.x == 0) ? 0u : sh[threadIdx.x - 1];
    #pragma unroll
    for (int i = 0; i < 4; ++i) blockSums[base + i] = tOff + local[i];
    if (threadIdx.x == SCAN_BLOCK - 1) *nUnique = sh[SCAN_BLOCK - 1];
}

__global__ void sc_scan3_kernel(unsigned* __restrict__ out,
                                const unsigned* __restrict__ blockSums) {
    unsigned v = blockSums[blockIdx.x];
    unsigned base = blockIdx.x * SCAN_ELEMS + threadIdx.x;
    #pragma unroll
    for (int i = 0; i < SCAN_ITEMS; ++i) out[base + i * SCAN_BLOCK] += v;
}

// ---------------------------------------------------------------------------
// 3) rank -> cell map
// ---------------------------------------------------------------------------
__global__ void sc_rankmap_kernel(const unsigned* __restrict__ flags,
                                  const unsigned* __restrict__ scanArr,
                                  unsigned* __restrict__ cellOfRank) {
    unsigned c = blockIdx.x * blockDim.x + threadIdx.x;
    if (c >= NCELLS) return;
    if (flags[c]) cellOfRank[scanArr[c]] = c;
}

// ---------------------------------------------------------------------------
// 4) segment-sum features into unique voxels (fds)
// ---------------------------------------------------------------------------
__global__ void sc_scatter_kernel(const unsigned* __restrict__ pointCell,
                                  const unsigned* __restrict__ scanArr,
                                  const float* __restrict__ feats,
                                  float* __restrict__ fds, int N) {
    long long t = (long long)blockIdx.x * blockDim.x + threadIdx.x;
    int i = (int)(t >> 6);
    int c = (int)(t & 63);
    if (i >= N) return;
    unsigned r = scanArr[pointCell[i]];
    atomicAdd(&fds[(size_t)r * CIN + c], feats[(size_t)i * CIN + c]);
}

// ---------------------------------------------------------------------------
// 5) gather + GEMM (fp32 WMMA 16x16x4) + BN + ReLU
//    block = 128 threads (4 waves); block tile = 16 rows x 64 cols,
//    each wave owns one 16x16 C tile.
// ---------------------------------------------------------------------------
__global__ __launch_bounds__(128) void sc_conv_kernel(
    const unsigned* __restrict__ cellOfRank,
    const unsigned* __restrict__ scanArr,
    const unsigned* __restrict__ flags,
    const float* __restrict__ fds,
    const float* __restrict__ weight,      // [27][64][64]
    const float* __restrict__ bn_scale,
    const float* __restrict__ bn_bias,
    const float* __restrict__ bn_mean,
    const float* __restrict__ bn_var,
    const unsigned* __restrict__ nUniquePtr,
    float* __restrict__ out) {

    __shared__ float Atile[16][CIN + 4];   // +4 pad: column reads hit distinct banks
    __shared__ int   nbRow[16];
    __shared__ int   cellInfo[16][4];      // b, x, y, z

    const unsigned nU = *nUniquePtr;
    const unsigned rowBase = blockIdx.x * 16;
    if (rowBase >= nU) return;             // out pre-zeroed; wave-uniform exit

    const int tid   = threadIdx.x;
    const int lane  = tid & 31;
    const int wave  = tid >> 5;            // 0..3 -> column tile
    const int m     = lane & 15;           // M index (A) / N index (B,C)
    const int khalf = lane >> 4;           // 0/1 -> K sub-slab
    const int col   = (wave << 4) + m;     // global output column 0..63

    if (tid < 16) {
        unsigned r = rowBase + (unsigned)tid;
        if (r < nU) {
            unsigned cell = cellOfRank[r];
            cellInfo[tid][0] = (int)(cell >> 21);
            cellInfo[tid][1] = (int)((cell >> 14) & 127u);
            cellInfo[tid][2] = (int)((cell >> 7) & 127u);
            cellInfo[tid][3] = (int)(cell & 127u);
        } else {
            cellInfo[tid][0] = 0;
            cellInfo[tid][1] = -1000;      // forces all neighbor lookups OOB
            cellInfo[tid][2] = 0;
            cellInfo[tid][3] = 0;
        }
    }
    __syncthreads();

    v8f acc = {0.f, 0.f, 0.f, 0.f, 0.f, 0.f, 0.f, 0.f};

    for (int k = 0; k < 27; ++k) {
        const int dx = k / 9 - 1;
        const int dy = (k / 3) % 3 - 1;
        const int dz = k % 3 - 1;

        if (tid < 16) {
            int x = cellInfo[tid][1] + dx;
            int y = cellInfo[tid][2] + dy;
            int z = cellInfo[tid][3] + dz;
            int nb = -1;
            if ((unsigned)x < 128u && (unsigned)y < 128u && (unsigned)z < 128u) {
                unsigned c = ((unsigned)cellInfo[tid][0] << 21) |
                             ((unsigned)x << 14) | ((unsigned)y << 7) | (unsigned)z;
                if (flags[c]) nb = (int)scanArr[c];
            }
            nbRow[tid] = nb;
        }
        __syncthreads();

        // stage gathered A tile: 16 rows x 64 ch, 8 floats per thread
        {
            const int r  = tid >> 3;
            const int c0 = (tid & 7) * 8;
            const int nb = nbRow[r];
            if (nb >= 0) {
                const float* src = fds + (size_t)nb * CIN + c0;
                #pragma unroll
                for (int j = 0; j < 8; ++j) Atile[r][c0 + j] = src[j];
            } else {
                #pragma unroll
                for (int j = 0; j < 8; ++j) Atile[r][c0 + j] = 0.f;
            }
        }
        __syncthreads();

        const float* Wk = weight + (size_t)k * CIN * COUT;
        #pragma unroll
        for (int kk = 0; kk < CIN; kk += 4) {
            const int kbase = kk + khalf * 2;
            v2f a, b;
            a.x = Atile[m][kbase];
            a.y = Atile[m][kbase + 1];
            b.x = Wk[(size_t)kbase * COUT + col];
            b.y = Wk[(size_t)(kbase + 1) * COUT + col];
            acc = __builtin_amdgcn_wmma_f32_16x16x4_f32(
                /*neg_a=*/false, a, /*neg_b=*/false, b,
                /*c_mod=*/(short)0, acc, /*reuse_a=*/false, /*reuse_b=*/false);
        }
        __syncthreads();
    }

    // fused BN + ReLU; C/D layout: elem i @ lane L -> M = i + 8*(L/16), N = L%16
    const float g  = bn_scale[col] * rsqrtf(bn_var[col] + 1e-5f);
    const float mu = bn_mean[col];
    const float be = bn_bias[col];
    #pragma unroll
    for (int i = 0; i < 8; ++i) {
        unsigned row = rowBase + (unsigned)(i + 8 * khalf);
        if (row < nU) {
            float v = (acc[i] - mu) * g + be;
            v = fmaxf(v, 0.f);
            out[(size_t)row * COUT + col] = v;
        }
    }
}

// ---------------------------------------------------------------------------
// launcher
// ---------------------------------------------------------------------------
extern "C" void kernel_launch(void* const* d_in, const int* in_sizes, int n_in,
                              void* d_out, int out_size, void* d_ws, size_t ws_size,
                              hipStream_t stream) {
    const int*   coords   = (const int*)d_in[0];
    const float* feats    = (const float*)d_in[1];
    const float* weight   = (const float*)d_in[2];
    const float* bn_scale = (const float*)d_in[3];
    const float* bn_bias  = (const float*)d_in[4];
    const float* bn_mean  = (const float*)d_in[5];
    const float* bn_var   = (const float*)d_in[6];
    float*       out      = (float*)d_out;

    const int N = in_sizes[0] / 4;   // coords is (N,4)

    // workspace carve-up (256B aligned)
    char* ws = (char*)d_ws;
    size_t off = 0;
    auto carve = [&](size_t bytes) -> void* {
        void* p = ws + off;
        off += (bytes + 255) & ~(size_t)255;
        return p;
    };
    unsigned* flags      = (unsigned*)carve((size_t)NCELLS * 4);
    unsigned* scanArr    = (unsigned*)carve((size_t)NCELLS * 4);
    unsigned* blockSums  = (unsigned*)carve((size_t)SCAN_NBLK * 4);
    unsigned* nUnique    = (unsigned*)carve(256);
    unsigned* pointCell  = (unsigned*)carve((size_t)N * 4);
    unsigned* cellOfRank = (unsigned*)carve((size_t)N * 4);
    float*    fds        = (float*)carve((size_t)N * CIN * 4);
    (void)ws_size;

    hipMemsetAsync(flags, 0, (size_t)NCELLS * 4, stream);
    hipMemsetAsync(fds,   0, (size_t)N * CIN * 4, stream);
    hipMemsetAsync(out,   0, (size_t)out_size * 4, stream);

    sc_keys_kernel<<<(N + 255) / 256, 256, 0, stream>>>(coords, flags, pointCell, N);

    sc_scan1_kernel<<<SCAN_NBLK, SCAN_BLOCK, 0, stream>>>(flags, scanArr, blockSums);
    sc_scan2_kernel<<<1, SCAN_BLOCK, 0, stream>>>(blockSums, nUnique);
    sc_scan3_kernel<<<SCAN_NBLK, SCAN_BLOCK, 0, stream>>>(scanArr, blockSums);

    sc_rankmap_kernel<<<NCELLS / 256, 256, 0, stream>>>(flags, scanArr, cellOfRank);

    {
        long long total = (long long)N * CIN;
        int blocks = (int)((total + 255) / 256);
        sc_scatter_kernel<<<blocks, 256, 0, stream>>>(pointCell, scanArr, feats, fds, N);
    }

    sc_conv_kernel<<<(N + 15) / 16, 128, 0, stream>>>(
        cellOfRank, scanArr, flags, fds, weight,
        bn_scale, bn_bias, bn_mean, bn_var, nUnique, out);
}